// TK_v3_42674795053882
// MI455X (gfx1250) — compile-verified
//
#include <hip/hip_runtime.h>

// ---------------------------------------------------------------------------
// TK ranking model for MI455X (gfx1250, wave32).
// GEMMs use V_WMMA_F32_16X16X4_F32 (fp32 matrix path, keeps reference
// precision). Each wave computes a 32x32 C block (2x2 WMMA tiles) with a
// software-pipelined K loop; edge tiles use clamp+select loads (no EXEC
// branching). Attention is fused (online softmax) so the 310MB scoring-stage
// logits are never materialized; the whole working set (~42MB) stays in L2.
// ---------------------------------------------------------------------------

#define BB 8
#define QL 30
#define DL 200
#define EMBD 300
#define M_LAYERS 2
#define M_HEADS 10
#define M_FF 100
#define S_LAYERS 3
#define S_HEADS 8
#define S_D 32
#define S_FF 32
#define S_SEQ (DL + 1)        // 201
#define S_BATCH (BB * QL)     // 240

typedef float v2f __attribute__((ext_vector_type(2)));
typedef float v8f __attribute__((ext_vector_type(8)));

#define WMMA_F32(A_, B_, C_) \
    __builtin_amdgcn_wmma_f32_16x16x4_f32(false, (A_), false, (B_), (short)0, (C_), false, false)

// ---------------------------------------------------------------------------
// Batched GEMM: C[b] = A[b](MxK) * B[b](KxN) (+bias) (+relu); transB => B is
// [N x K] row-major. Requires K % 4 == 0 (true for all call sites: 300/100/32).
// Block = 128 threads (4 waves); wave tile = 32x32 (2x2 WMMA tiles).
// WMMA VGPR layout (ISA 7.12.2, 32-bit, wave32):
//   A 16x4 : lanes 0-15 hold {K=0,K=1}, lanes 16-31 hold {K=2,K=3}, M = lane%16
//   B 4x16 : same K split, N = lane%16
//   C 16x16: VGPR i -> M = i + 8*(lane/16), N = lane%16
// ---------------------------------------------------------------------------
__global__ __launch_bounds__(128)
void tk_gemm_wmma_f32(const float* __restrict__ A, const float* __restrict__ Bm,
                      const float* __restrict__ bias, float* __restrict__ C,
                      int M, int N, int K, int lda, int ldb, int ldc,
                      long sA, long sB, long sC, int transB, int relu)
{
    const int batch = blockIdx.z;
    A  += (long)batch * sA;
    Bm += (long)batch * sB;
    C  += (long)batch * sC;

    const int lane = threadIdx.x & 31;
    const int wave = threadIdx.x >> 5;
    const int n0 = blockIdx.x * 32;
    const int m0 = (blockIdx.y * 4 + wave) * 32;
    if (m0 >= M || n0 >= N) return;   // wave-uniform: EXEC stays all-ones

    const int half = lane >> 4;       // 0: K pair {0,1}   1: K pair {2,3}
    const int l    = lane & 15;
    const int am0  = m0 + l;
    const int am1  = m0 + 16 + l;
    const int bn0  = n0 + l;
    const int bn1  = n0 + 16 + l;

    v8f acc00 = {0.f, 0.f, 0.f, 0.f, 0.f, 0.f, 0.f, 0.f};
    v8f acc01 = acc00, acc10 = acc00, acc11 = acc00;

    const bool full = (m0 + 32 <= M) && (n0 + 32 <= N);
    if (full) {
        const float* ap0 = A + (long)am0 * lda + half * 2;
        const float* ap1 = A + (long)am1 * lda + half * 2;
        // prologue: load k-step 0
        v2f a0, a1, b0, b1;
        {
            const int ka = half * 2;
            a0.x = ap0[0]; a0.y = ap0[1];
            a1.x = ap1[0]; a1.y = ap1[1];
            if (!transB) {
                b0.x = Bm[(long)ka * ldb + bn0]; b0.y = Bm[(long)(ka + 1) * ldb + bn0];
                b1.x = Bm[(long)ka * ldb + bn1]; b1.y = Bm[(long)(ka + 1) * ldb + bn1];
            } else {
                b0.x = Bm[(long)bn0 * ldb + ka]; b0.y = Bm[(long)bn0 * ldb + ka + 1];
                b1.x = Bm[(long)bn1 * ldb + ka]; b1.y = Bm[(long)bn1 * ldb + ka + 1];
            }
        }
        // software-pipelined main loop: issue next loads before current wmmas
        for (int k0 = 0; k0 < K; k0 += 4) {
            v2f na0 = a0, na1 = a1, nb0 = b0, nb1 = b1;
            const int kn = k0 + 4;
            if (kn < K) {                       // uniform branch (no EXEC churn)
                const int ka = kn + half * 2;
                na0.x = ap0[kn]; na0.y = ap0[kn + 1];
                na1.x = ap1[kn]; na1.y = ap1[kn + 1];
                if (!transB) {
                    nb0.x = Bm[(long)ka * ldb + bn0]; nb0.y = Bm[(long)(ka + 1) * ldb + bn0];
                    nb1.x = Bm[(long)ka * ldb + bn1]; nb1.y = Bm[(long)(ka + 1) * ldb + bn1];
                } else {
                    nb0.x = Bm[(long)bn0 * ldb + ka]; nb0.y = Bm[(long)bn0 * ldb + ka + 1];
                    nb1.x = Bm[(long)bn1 * ldb + ka]; nb1.y = Bm[(long)bn1 * ldb + ka + 1];
                }
            }
            acc00 = WMMA_F32(a0, b0, acc00);
            acc01 = WMMA_F32(a0, b1, acc01);
            acc10 = WMMA_F32(a1, b0, acc10);
            acc11 = WMMA_F32(a1, b1, acc11);
            a0 = na0; a1 = na1; b0 = nb0; b1 = nb1;
        }
    } else {
        // edge tiles: unconditional loads from clamped addresses + select 0
        const int amc0 = min(am0, M - 1);
        const int amc1 = min(am1, M - 1);
        const int bnc0 = min(bn0, N - 1);
        const int bnc1 = min(bn1, N - 1);
        const bool va0 = (am0 < M), va1 = (am1 < M);
        const bool vb0 = (bn0 < N), vb1 = (bn1 < N);
        for (int k0 = 0; k0 < K; k0 += 4) {
            const int ka = k0 + half * 2;
            v2f a0, a1, b0, b1;
            float t;
            t = A[(long)amc0 * lda + ka];     a0.x = va0 ? t : 0.f;
            t = A[(long)amc0 * lda + ka + 1]; a0.y = va0 ? t : 0.f;
            t = A[(long)amc1 * lda + ka];     a1.x = va1 ? t : 0.f;
            t = A[(long)amc1 * lda + ka + 1]; a1.y = va1 ? t : 0.f;
            if (!transB) {
                t = Bm[(long)ka * ldb + bnc0];       b0.x = vb0 ? t : 0.f;
                t = Bm[(long)(ka + 1) * ldb + bnc0]; b0.y = vb0 ? t : 0.f;
                t = Bm[(long)ka * ldb + bnc1];       b1.x = vb1 ? t : 0.f;
                t = Bm[(long)(ka + 1) * ldb + bnc1]; b1.y = vb1 ? t : 0.f;
            } else {
                t = Bm[(long)bnc0 * ldb + ka];     b0.x = vb0 ? t : 0.f;
                t = Bm[(long)bnc0 * ldb + ka + 1]; b0.y = vb0 ? t : 0.f;
                t = Bm[(long)bnc1 * ldb + ka];     b1.x = vb1 ? t : 0.f;
                t = Bm[(long)bnc1 * ldb + ka + 1]; b1.y = vb1 ? t : 0.f;
            }
            acc00 = WMMA_F32(a0, b0, acc00);
            acc01 = WMMA_F32(a0, b1, acc01);
            acc10 = WMMA_F32(a1, b0, acc10);
            acc11 = WMMA_F32(a1, b1, acc11);
        }
    }

    // epilogue: 2x2 tiles, bias/relu, bounds-checked stores
    const int cn0 = n0 + l;
    const int cn1 = n0 + 16 + l;
    const int cm0 = m0 + half * 8;
    const int cm1 = m0 + 16 + half * 8;
    const float bv0 = bias ? bias[min(cn0, N - 1)] : 0.f;
    const float bv1 = bias ? bias[min(cn1, N - 1)] : 0.f;
#pragma unroll
    for (int i = 0; i < 8; ++i) {
        float v;
        if (cm0 + i < M && cn0 < N) {
            v = acc00[i] + bv0; if (relu) v = fmaxf(v, 0.f);
            C[(long)(cm0 + i) * ldc + cn0] = v;
        }
        if (cm0 + i < M && cn1 < N) {
            v = acc01[i] + bv1; if (relu) v = fmaxf(v, 0.f);
            C[(long)(cm0 + i) * ldc + cn1] = v;
        }
        if (cm1 + i < M && cn0 < N) {
            v = acc10[i] + bv0; if (relu) v = fmaxf(v, 0.f);
            C[(long)(cm1 + i) * ldc + cn0] = v;
        }
        if (cm1 + i < M && cn1 < N) {
            v = acc11[i] + bv1; if (relu) v = fmaxf(v, 0.f);
            C[(long)(cm1 + i) * ldc + cn1] = v;
        }
    }
}

// ---------------------------------------------------------------------------
// Fused multi-head attention with key padding mask + online softmax.
// One thread = one query row; grid = (ceil(S/256), B, heads).
// Matches reference: s = (q.k)/sqrt(dh); s = mask>0 ? s : -1e9; softmax; P@V.
// ---------------------------------------------------------------------------
template <int DH>
__global__ __launch_bounds__(256)
void tk_attention(const float* __restrict__ Q, const float* __restrict__ K,
                  const float* __restrict__ V, const float* __restrict__ mask,
                  float* __restrict__ O, int S, int D)
{
    const int b = blockIdx.y;
    const int h = blockIdx.z;
    const int q = blockIdx.x * blockDim.x + threadIdx.x;
    if (q >= S) return;

    const float scale = rsqrtf((float)DH);
    const long rowb = (long)b * S;
    const float* qp = Q + (rowb + q) * D + h * DH;

    float qreg[DH];
#pragma unroll
    for (int i = 0; i < DH; ++i) qreg[i] = qp[i];

    float m = -3.0e38f, lsum = 0.f;
    float acc[DH];
#pragma unroll
    for (int i = 0; i < DH; ++i) acc[i] = 0.f;

    const float* mrow = mask + rowb;
    for (int k = 0; k < S; ++k) {
        const float* kp = K + (rowb + k) * D + h * DH;
        float s = 0.f;
#pragma unroll
        for (int i = 0; i < DH; ++i) s += qreg[i] * kp[i];
        s *= scale;
        if (!(mrow[k] > 0.f)) s = -1e9f;
        const float mn   = fmaxf(m, s);
        const float corr = __expf(m - mn);
        const float w    = __expf(s - mn);
        lsum = lsum * corr + w;
        const float* vp = V + (rowb + k) * D + h * DH;
#pragma unroll
        for (int i = 0; i < DH; ++i) acc[i] = acc[i] * corr + w * vp[i];
        m = mn;
    }
    const float inv = 1.f / lsum;
    float* op = O + (rowb + q) * D + h * DH;
#pragma unroll
    for (int i = 0; i < DH; ++i) op[i] = acc[i] * inv;
}

// x = LayerNorm(x + y) * g + b   (post-LN, biased variance, eps=1e-5)
__global__ __launch_bounds__(128)
void tk_add_ln(float* __restrict__ x, const float* __restrict__ y,
               const float* __restrict__ g, const float* __restrict__ bb, int D)
{
    const long row = blockIdx.x;
    float* xr = x + row * D;
    const float* yr = y + row * D;
    __shared__ float s1[128], s2[128];
    float sum = 0.f, sq = 0.f;
    for (int i = threadIdx.x; i < D; i += 128) {
        const float t = xr[i] + yr[i];
        sum += t; sq += t * t;
    }
    s1[threadIdx.x] = sum; s2[threadIdx.x] = sq;
    __syncthreads();
    for (int o = 64; o > 0; o >>= 1) {
        if (threadIdx.x < o) {
            s1[threadIdx.x] += s1[threadIdx.x + o];
            s2[threadIdx.x] += s2[threadIdx.x + o];
        }
        __syncthreads();
    }
    const float mean = s1[0] / D;
    const float var  = s2[0] / D - mean * mean;
    const float rstd = rsqrtf(var + 1e-5f);
    for (int i = threadIdx.x; i < D; i += 128) {
        const float t = xr[i] + yr[i];
        xr[i] = (t - mean) * rstd * g[i] + bb[i];
    }
}

// row-wise L2 normalize: x /= (||x|| + 1e-13)   (allennlp cosine epsilon)
__global__ __launch_bounds__(128)
void tk_rownorm(float* __restrict__ x, int D)
{
    const long row = blockIdx.x;
    float* xr = x + row * D;
    __shared__ float s1[128];
    float sq = 0.f;
    for (int i = threadIdx.x; i < D; i += 128) { const float t = xr[i]; sq += t * t; }
    s1[threadIdx.x] = sq;
    __syncthreads();
    for (int o = 64; o > 0; o >>= 1) {
        if (threadIdx.x < o) s1[threadIdx.x] += s1[threadIdx.x + o];
        __syncthreads();
    }
    const float inv = 1.f / (sqrtf(s1[0]) + 1e-13f);
    for (int i = threadIdx.x; i < D; i += 128) xr[i] *= inv;
}

// o1 = o2 = e * mask[row]   (row = blockIdx.y; no integer division)
__global__ void tk_maskmul2(const float* __restrict__ e, const float* __restrict__ m,
                            float* __restrict__ o1, float* __restrict__ o2, int D)
{
    const int row = blockIdx.y;
    const int i = blockIdx.x * blockDim.x + threadIdx.x;
    if (i >= D) return;
    const long idx = (long)row * D + i;
    const float v = e[idx] * m[row];
    o1[idx] = v;
    o2[idx] = v;
}

// out = (mix * e0 + (1-mix) * ctx) * mask[row]
__global__ void tk_mix(const float* __restrict__ e0, const float* __restrict__ ctx,
                       const float* __restrict__ m, const float* __restrict__ mixer,
                       float* __restrict__ o, int D)
{
    const int row = blockIdx.y;
    const int i = blockIdx.x * blockDim.x + threadIdx.x;
    if (i >= D) return;
    const long idx = (long)row * D + i;
    const float mx = mixer[0];
    o[idx] = (mx * e0[idx] + (1.f - mx) * ctx[idx]) * m[row];
}

// Build scoring-stage input tokens:
// token(pos=0) = cls ; token(pos>0) = cosine * qm*dm ; then scalar @ Win + bin
// + sinusoidal posenc. Also writes the scoring key mask.
__global__ void tk_build_scoring(const float* __restrict__ cosine,
                                 const float* __restrict__ qm, const float* __restrict__ dm,
                                 const float* __restrict__ cls, const float* __restrict__ Win,
                                 const float* __restrict__ bin, float* __restrict__ xs,
                                 float* __restrict__ ms)
{
    const long idx = (long)blockIdx.x * blockDim.x + threadIdx.x;
    const long total = (long)S_BATCH * S_SEQ * S_D;
    if (idx >= total) return;
    const int j  = (int)(idx & (S_D - 1));
    const long t = idx >> 5;                 // token index (S_D == 32)
    const int pos = (int)(t % S_SEQ);
    const long bq = t / S_SEQ;
    const int qq = (int)(bq % QL);
    const int b  = (int)(bq / QL);

    float mval, tval;
    if (pos == 0) { mval = 1.f; tval = cls[0]; }
    else {
        mval = qm[b * QL + qq] * dm[b * DL + pos - 1];
        tval = cosine[((long)(b * QL + qq)) * DL + (pos - 1)] * mval;
    }
    const int jj = j & ~1;
    const float ang = (float)pos * __powf(10000.f, -(float)jj / (float)S_D);
    const float pe  = (j & 1) ? __cosf(ang) : __sinf(ang);
    xs[idx] = tval * Win[j] + bin[j] + pe;
    if (j == 0) ms[t] = mval;
}

// out[b] = sum_q (CLS[b,q] . funnel_w) * qm / sum_q qm
__global__ __launch_bounds__(64)
void tk_final(const float* __restrict__ ys, const float* __restrict__ fw,
              const float* __restrict__ qm, float* __restrict__ out)
{
    const int b = blockIdx.x;
    __shared__ float s1[64], s2[64];
    float sc = 0.f, sm = 0.f;
    for (int q = threadIdx.x; q < QL; q += 64) {
        const float* rep = ys + ((long)(b * QL + q) * S_SEQ) * S_D;
        float dot = 0.f;
        for (int j = 0; j < S_D; ++j) dot += rep[j] * fw[j];
        const float mv = qm[b * QL + q];
        sc += dot * mv; sm += mv;
    }
    s1[threadIdx.x] = sc; s2[threadIdx.x] = sm;
    __syncthreads();
    for (int o = 32; o > 0; o >>= 1) {
        if (threadIdx.x < o) {
            s1[threadIdx.x] += s1[threadIdx.x + o];
            s2[threadIdx.x] += s2[threadIdx.x + o];
        }
        __syncthreads();
    }
    if (threadIdx.x == 0) out[b] = s1[0] / s2[0];
}

// ---------------------------------------------------------------------------
// Host orchestration
// ---------------------------------------------------------------------------
struct EncP {
    const float *Wq, *Wk, *Wv, *Wo, *bq, *bk, *bv, *bo;
    const float *W1, *b1, *W2, *b2, *l1g, *l1b, *l2g, *l2b;
};

extern "C" void kernel_launch(void* const* d_in, const int* in_sizes, int n_in,
                              void* d_out, int out_size, void* d_ws, size_t ws_size,
                              hipStream_t stream)
{
    (void)n_in; (void)out_size; (void)ws_size;
    auto F = [&](int i) { return (const float*)d_in[i]; };

    const float* qe    = F(0);
    const float* de    = F(1);
    const float* qm    = F(2);
    const float* dm    = F(3);
    const float* mixer = F(4);
    const float* cls   = F(5);
    const float* fw    = F(6);

    // Resolve nested-dict flatten order at runtime:
    //   insertion order  -> in_sizes[7] == 2*300*300 (Wq)
    //   sorted-key order -> in_sizes[7] == 2*300*100 (W1)
    EncP mp, sp;
    const float *Win, *binp;
    const bool sorted = (in_sizes[7] == M_LAYERS * EMBD * M_FF);
    if (!sorted) {
        mp = { F(7), F(8), F(9), F(10), F(11), F(12), F(13), F(14),
               F(15), F(16), F(17), F(18), F(19), F(20), F(21), F(22) };
        sp = { F(23), F(24), F(25), F(26), F(27), F(28), F(29), F(30),
               F(31), F(32), F(33), F(34), F(35), F(36), F(37), F(38) };
        Win = F(39); binp = F(40);
    } else {
        mp.W1 = F(7);  mp.W2 = F(8);  mp.Wk = F(9);  mp.Wo = F(10);
        mp.Wq = F(11); mp.Wv = F(12); mp.b1 = F(13); mp.b2 = F(14);
        mp.bk = F(15); mp.bo = F(16); mp.bq = F(17); mp.bv = F(18);
        mp.l1b = F(19); mp.l1g = F(20); mp.l2b = F(21); mp.l2g = F(22);
        sp.W1 = F(23); sp.W2 = F(24); Win = F(25); sp.Wk = F(26);
        sp.Wo = F(27); sp.Wq = F(28); sp.Wv = F(29); sp.b1 = F(30);
        sp.b2 = F(31); binp = F(32); sp.bk = F(33); sp.bo = F(34);
        sp.bq = F(35); sp.bv = F(36); sp.l1b = F(37); sp.l1g = F(38);
        sp.l2b = F(39); sp.l2g = F(40);
    }

    // Workspace carve (~42 MB of f32)
    float* w = (float*)d_ws;
    auto alloc = [&](long n) { float* p = w; w += n; return p; };
    const long NQ  = (long)BB * QL * EMBD;            //  72000
    const long ND  = (long)BB * DL * EMBD;            // 480000
    const long BIG = (long)S_BATCH * S_SEQ * S_D;     // 1543680
    float* eq0 = alloc(NQ);     // masked raw query embeddings
    float* ed0 = alloc(ND);     // masked raw doc embeddings
    float* xq  = alloc(NQ);     // query activations
    float* xd  = alloc(ND);     // doc activations
    float* Qb  = alloc(BIG);
    float* Kb  = alloc(BIG);
    float* Vb  = alloc(BIG);
    float* Ob  = alloc(BIG);
    float* Hf  = alloc(BIG);
    float* cosb = alloc((long)BB * QL * DL);          // 48000
    float* xs  = alloc(BIG);    // scoring-stage activations
    float* ms  = alloc((long)S_BATCH * S_SEQ);        // scoring key mask

    auto gemm = [&](const float* A, const float* Bm, const float* bias, float* C,
                    int M, int N, int K, int lda, int ldb, int ldc,
                    int batch, long sA, long sB, long sC, int transB, int relu) {
        dim3 grid((N + 31) / 32, (M + 127) / 128, batch);
        tk_gemm_wmma_f32<<<grid, dim3(128), 0, stream>>>(
            A, Bm, bias, C, M, N, K, lda, ldb, ldc, sA, sB, sC, transB, relu);
    };

    auto encoder = [&](float* X, const float* mask, int Bn, int S, int D, int ff,
                       int heads, int nlayers, const EncP& p) {
        const int Mr = Bn * S;
        const int dh = D / heads;
        for (int l = 0; l < nlayers; ++l) {
            const float* Wq = p.Wq + (long)l * D * D;
            const float* Wk = p.Wk + (long)l * D * D;
            const float* Wv = p.Wv + (long)l * D * D;
            const float* Wo = p.Wo + (long)l * D * D;
            const float* bq = p.bq + (long)l * D;
            const float* bk = p.bk + (long)l * D;
            const float* bv = p.bv + (long)l * D;
            const float* bo = p.bo + (long)l * D;
            const float* W1 = p.W1 + (long)l * D * ff;
            const float* b1 = p.b1 + (long)l * ff;
            const float* W2 = p.W2 + (long)l * ff * D;
            const float* b2 = p.b2 + (long)l * D;
            const float* l1g = p.l1g + (long)l * D;
            const float* l1b = p.l1b + (long)l * D;
            const float* l2g = p.l2g + (long)l * D;
            const float* l2b = p.l2b + (long)l * D;

            // QKV projections
            gemm(X, Wq, bq, Qb, Mr, D, D, D, D, D, 1, 0, 0, 0, 0, 0);
            gemm(X, Wk, bk, Kb, Mr, D, D, D, D, D, 1, 0, 0, 0, 0, 0);
            gemm(X, Wv, bv, Vb, Mr, D, D, D, D, D, 1, 0, 0, 0, 0, 0);
            // fused attention
            dim3 ag((S + 255) / 256, Bn, heads);
            if (dh == 30)
                tk_attention<30><<<ag, 256, 0, stream>>>(Qb, Kb, Vb, mask, Ob, S, D);
            else
                tk_attention<4><<<ag, 256, 0, stream>>>(Qb, Kb, Vb, mask, Ob, S, D);
            // output projection + residual LN
            gemm(Ob, Wo, bo, Hf, Mr, D, D, D, D, D, 1, 0, 0, 0, 0, 0);
            tk_add_ln<<<Mr, 128, 0, stream>>>(X, Hf, l1g, l1b, D);
            // FFN + residual LN
            gemm(X, W1, b1, Qb, Mr, ff, D, D, ff, ff, 1, 0, 0, 0, 0, 1);
            gemm(Qb, W2, b2, Hf, Mr, D, ff, ff, D, D, 1, 0, 0, 0, 0, 0);
            tk_add_ln<<<Mr, 128, 0, stream>>>(X, Hf, l2g, l2b, D);
        }
    };

    // Stage 1: mask raw embeddings (dual write: e0 copy + working activation)
    {
        dim3 gq((EMBD + 255) / 256, BB * QL);
        dim3 gd((EMBD + 255) / 256, BB * DL);
        tk_maskmul2<<<gq, 256, 0, stream>>>(qe, qm, eq0, xq, EMBD);
        tk_maskmul2<<<gd, 256, 0, stream>>>(de, dm, ed0, xd, EMBD);
    }
    encoder(xq, qm, BB, QL, EMBD, M_FF, M_HEADS, M_LAYERS, mp);
    encoder(xd, dm, BB, DL, EMBD, M_FF, M_HEADS, M_LAYERS, mp);

    // Stage 2: mix, L2-normalize, batched cosine matrix (transB WMMA GEMM)
    {
        dim3 gq((EMBD + 255) / 256, BB * QL);
        dim3 gd((EMBD + 255) / 256, BB * DL);
        tk_mix<<<gq, 256, 0, stream>>>(eq0, xq, qm, mixer, Qb, EMBD);
        tk_mix<<<gd, 256, 0, stream>>>(ed0, xd, dm, mixer, Kb, EMBD);
    }
    tk_rownorm<<<BB * QL, 128, 0, stream>>>(Qb, EMBD);
    tk_rownorm<<<BB * DL, 128, 0, stream>>>(Kb, EMBD);
    gemm(Qb, Kb, nullptr, cosb, QL, DL, EMBD, EMBD, EMBD, DL,
         BB, (long)QL * EMBD, (long)DL * EMBD, (long)QL * DL, /*transB=*/1, 0);

    // Stage 3: build scoring tokens (cls + cosine, scalar->32 proj + posenc)
    tk_build_scoring<<<(unsigned)((BIG + 255) / 256), 256, 0, stream>>>(
        cosb, qm, dm, cls, Win, binp, xs, ms);

    // Stage 4: scoring encoder (240 x 201 x 32)
    encoder(xs, ms, S_BATCH, S_SEQ, S_D, S_FF, S_HEADS, S_LAYERS, sp);

    // Stage 5: CLS -> funnel -> masked mean
    tk_final<<<BB, 64, 0, stream>>>(xs, fw, qm, (float*)d_out);
}